// VariantSAGEGNN_8830452761021
// MI455X (gfx1250) — compile-verified
//
#include <hip/hip_runtime.h>

// ---------------------------------------------------------------------------
// GraphSAGE (3 blocks) for MI455X / gfx1250.
//   N=100000 nodes, E=1600000 edges, H=D=128, C=2, fp32 throughout.
//   GEMMs use V_WMMA_F32_16X16X4_F32 (exact fp32 matrix path, wave32).
//   Workspace requirement: 3 * 51.2MB (h ping-pong + agg) + ~1MB small = ~155MB.
// ---------------------------------------------------------------------------

#define N_NODES 100000
#define N_EDGES 1600000
#define DIM     128
#define EPSV    1e-5f

typedef __attribute__((ext_vector_type(2))) float v2f;
typedef __attribute__((ext_vector_type(8))) float v8f;

__device__ __forceinline__ v8f wmma_f32x4(v2f a, v2f b, v8f c) {
  // D = A(16x4 f32) * B(4x16 f32) + C(16x16 f32)
  return __builtin_amdgcn_wmma_f32_16x16x4_f32(
      /*neg_a=*/false, a, /*neg_b=*/false, b,
      /*c_mod=*/(short)0, c, /*reuse_a=*/false, /*reuse_b=*/false);
}

// ---------------------------------------------------------------------------
// Utility: zero a buffer (float4 granularity)
// ---------------------------------------------------------------------------
__global__ void zero4_kernel(float4* __restrict__ p, int n4) {
  int i = blockIdx.x * blockDim.x + threadIdx.x;
  if (i < n4) p[i] = make_float4(0.f, 0.f, 0.f, 0.f);
}

// ---------------------------------------------------------------------------
// Degree: one thread per edge, count into deg[dst]
// ---------------------------------------------------------------------------
__global__ void deg_kernel(const int* __restrict__ ei, float* __restrict__ deg) {
  int e = blockIdx.x * blockDim.x + threadIdx.x;
  if (e < N_EDGES) atomicAdd(&deg[ei[N_EDGES + e]], 1.0f);
}

__global__ void invdeg_kernel(const float* __restrict__ deg, float* __restrict__ inv) {
  int i = blockIdx.x * blockDim.x + threadIdx.x;
  if (i < N_NODES) inv[i] = 1.0f / fmaxf(deg[i], 1.0f);
}

// ---------------------------------------------------------------------------
// Scatter: one wave per edge; lane handles 4 consecutive features.
// Wave reads h[src] row (512B coalesced), atomic-adds into agg[dst] row.
// Grid is exact: E*32 threads = 51.2M = 200000 blocks * 256.
// ---------------------------------------------------------------------------
__global__ void scatter_kernel(const float* __restrict__ h,
                               const int* __restrict__ ei,
                               float* __restrict__ agg) {
  int gid = blockIdx.x * 256 + threadIdx.x;
  int e = gid >> 5;
  int c = (gid & 31) << 2;
  int src = ei[e];
  int dst = ei[N_EDGES + e];
  const float4 v = *(const float4*)(h + (size_t)src * DIM + c);
  float* a = agg + (size_t)dst * DIM + c;
  atomicAdd(a + 0, v.x);
  atomicAdd(a + 1, v.y);
  atomicAdd(a + 2, v.z);
  atomicAdd(a + 3, v.w);
}

// ---------------------------------------------------------------------------
// Input GEMM: H = relu(X @ W + B).  X:[N,128] W:[128,128] row-major.
// Block = 16 output rows; 8 waves, wave w owns columns [16w,16w+16).
// ---------------------------------------------------------------------------
__global__ void gemm_in_kernel(const float* __restrict__ X,
                               const float* __restrict__ W,
                               const float* __restrict__ B,
                               float* __restrict__ H) {
  const int lane = threadIdx.x & 31;
  const int wave = threadIdx.x >> 5;
  const int half = lane >> 4;
  const int l16  = lane & 15;
  const int row  = blockIdx.x * 16;
  const int col  = wave * 16 + l16;

  v8f acc = {};
  const float* arow = X + (size_t)(row + l16) * DIM + 2 * half;
  const float* bcol = W + (size_t)(2 * half) * DIM + col;
#pragma unroll
  for (int k = 0; k < DIM; k += 4) {
    v2f a; a.x = arow[k];            a.y = arow[k + 1];
    v2f b; b.x = bcol[(size_t)k * DIM]; b.y = bcol[(size_t)k * DIM + DIM];
    acc = wmma_f32x4(a, b, acc);
  }
  const float bias = B[col];
  float* drow = H + (size_t)(row + 8 * half) * DIM + col;
#pragma unroll
  for (int r = 0; r < 8; ++r)
    drow[(size_t)r * DIM] = fmaxf(acc[r] + bias, 0.0f);
}

// ---------------------------------------------------------------------------
// SAGE GEMM: out = (agg * invdeg) @ LW + LB + h @ RW
// Also accumulates per-column sum / sumsq (pre-BN) into stats[0..127]/[128..255]
// via LDS reduction + one global atomicAdd per column per block.
// ---------------------------------------------------------------------------
__global__ void gemm_sage_kernel(const float* __restrict__ agg,
                                 const float* __restrict__ invdeg,
                                 const float* __restrict__ hin,
                                 const float* __restrict__ LW,
                                 const float* __restrict__ LB,
                                 const float* __restrict__ RW,
                                 float* __restrict__ outp,
                                 float* __restrict__ stats) {
  __shared__ float s_sum[DIM];
  __shared__ float s_sq[DIM];
  const int t = threadIdx.x;
  if (t < DIM) { s_sum[t] = 0.f; s_sq[t] = 0.f; }
  __syncthreads();

  const int lane = t & 31;
  const int wave = t >> 5;
  const int half = lane >> 4;
  const int l16  = lane & 15;
  const int row  = blockIdx.x * 16;
  const int col  = wave * 16 + l16;

  const float invd = invdeg[row + l16];
  v8f acc = {};

  // mean @ LW (mean formed on the fly: agg * invdeg)
  {
    const float* arow = agg + (size_t)(row + l16) * DIM + 2 * half;
    const float* bcol = LW + (size_t)(2 * half) * DIM + col;
#pragma unroll
    for (int k = 0; k < DIM; k += 4) {
      v2f a; a.x = arow[k] * invd;      a.y = arow[k + 1] * invd;
      v2f b; b.x = bcol[(size_t)k * DIM]; b.y = bcol[(size_t)k * DIM + DIM];
      acc = wmma_f32x4(a, b, acc);
    }
  }
  // h @ RW
  {
    const float* arow = hin + (size_t)(row + l16) * DIM + 2 * half;
    const float* bcol = RW + (size_t)(2 * half) * DIM + col;
#pragma unroll
    for (int k = 0; k < DIM; k += 4) {
      v2f a; a.x = arow[k];             a.y = arow[k + 1];
      v2f b; b.x = bcol[(size_t)k * DIM]; b.y = bcol[(size_t)k * DIM + DIM];
      acc = wmma_f32x4(a, b, acc);
    }
  }

  const float bias = LB[col];
  float csum = 0.f, csq = 0.f;
  float* drow = outp + (size_t)(row + 8 * half) * DIM + col;
#pragma unroll
  for (int r = 0; r < 8; ++r) {
    float v = acc[r] + bias;
    drow[(size_t)r * DIM] = v;   // pre-BN value; BN applied after stats pass
    csum += v;
    csq  += v * v;
  }
  atomicAdd(&s_sum[col], csum);  // ds_add_f32 (2 lanes per column per wave)
  atomicAdd(&s_sq[col],  csq);
  __syncthreads();
  if (t < DIM) {
    atomicAdd(&stats[t],       s_sum[t]);
    atomicAdd(&stats[DIM + t], s_sq[t]);
  }
}

// ---------------------------------------------------------------------------
// Fold BN (batch stats, biased var) + gamma/beta into scale/shift per feature
// ---------------------------------------------------------------------------
__global__ void bn_finalize_kernel(const float* __restrict__ stats,
                                   const float* __restrict__ g,
                                   const float* __restrict__ bt,
                                   float* __restrict__ ss) {
  int t = threadIdx.x;  // 128 threads
  float mu  = stats[t] * (1.0f / (float)N_NODES);
  float var = stats[DIM + t] * (1.0f / (float)N_NODES) - mu * mu;
  float sc  = rsqrtf(var + EPSV) * g[t];
  ss[t]       = sc;
  ss[DIM + t] = bt[t] - mu * sc;
}

// ---------------------------------------------------------------------------
// Apply BN + relu + residual, in place on `out`:
//   out = relu(out*scale + shift) + hin
// ---------------------------------------------------------------------------
__global__ void bn_apply_kernel(float* __restrict__ out,
                                const float* __restrict__ hin,
                                const float* __restrict__ ss) {
  int i = blockIdx.x * 256 + threadIdx.x;   // exact: N*DIM/4 threads
  int c = (i << 2) & (DIM - 1);
  float4 v  = ((float4*)out)[i];
  float4 h  = ((const float4*)hin)[i];
  float4 sc = *(const float4*)(ss + c);
  float4 sh = *(const float4*)(ss + DIM + c);
  v.x = fmaxf(v.x * sc.x + sh.x, 0.f) + h.x;
  v.y = fmaxf(v.y * sc.y + sh.y, 0.f) + h.y;
  v.z = fmaxf(v.z * sc.z + sh.z, 0.f) + h.z;
  v.w = fmaxf(v.w * sc.w + sh.w, 0.f) + h.w;
  ((float4*)out)[i] = v;
}

// ---------------------------------------------------------------------------
// Classifier: logits[n] = h[n] @ cls_w + cls_b  (128 -> 2, too thin for WMMA)
// ---------------------------------------------------------------------------
__global__ void cls_kernel(const float* __restrict__ h,
                           const float* __restrict__ w,
                           const float* __restrict__ b,
                           float* __restrict__ out) {
  int n = blockIdx.x * blockDim.x + threadIdx.x;
  if (n >= N_NODES) return;
  const float* row = h + (size_t)n * DIM;
  float a0 = 0.f, a1 = 0.f;
#pragma unroll
  for (int k = 0; k < DIM; k += 4) {
    float4 x = *(const float4*)(row + k);
    a0 += x.x * w[(k + 0) * 2 + 0] + x.y * w[(k + 1) * 2 + 0] +
          x.z * w[(k + 2) * 2 + 0] + x.w * w[(k + 3) * 2 + 0];
    a1 += x.x * w[(k + 0) * 2 + 1] + x.y * w[(k + 1) * 2 + 1] +
          x.z * w[(k + 2) * 2 + 1] + x.w * w[(k + 3) * 2 + 1];
  }
  out[(size_t)n * 2 + 0] = a0 + b[0];
  out[(size_t)n * 2 + 1] = a1 + b[1];
}

// ---------------------------------------------------------------------------
extern "C" void kernel_launch(void* const* d_in, const int* in_sizes, int n_in,
                              void* d_out, int out_size, void* d_ws, size_t ws_size,
                              hipStream_t stream) {
  (void)in_sizes; (void)n_in; (void)out_size; (void)ws_size;

  const float* x    = (const float*)d_in[0];
  const int*   ei   = (const int*)d_in[1];
  const float* in_w = (const float*)d_in[2];
  const float* in_b = (const float*)d_in[3];
  const float* cw   = (const float*)d_in[4];
  const float* cb   = (const float*)d_in[5];
  const float* lw[3] = {(const float*)d_in[6],  (const float*)d_in[11], (const float*)d_in[16]};
  const float* lb[3] = {(const float*)d_in[7],  (const float*)d_in[12], (const float*)d_in[17]};
  const float* rw[3] = {(const float*)d_in[8],  (const float*)d_in[13], (const float*)d_in[18]};
  const float* gg[3] = {(const float*)d_in[9],  (const float*)d_in[14], (const float*)d_in[19]};
  const float* bt[3] = {(const float*)d_in[10], (const float*)d_in[15], (const float*)d_in[20]};

  // Workspace layout
  const size_t MATB = (size_t)N_NODES * DIM * sizeof(float);  // 51.2 MB
  char* ws = (char*)d_ws;
  float* bufA  = (float*)(ws);
  float* bufB  = (float*)(ws + MATB);
  float* agg   = (float*)(ws + 2 * MATB);
  float* deg   = (float*)(ws + 3 * MATB);
  float* invd  = (float*)(ws + 3 * MATB + (size_t)(512 * 1024));
  float* stats = (float*)(ws + 3 * MATB + (size_t)(1024 * 1024));
  float* ss    = stats + 2 * DIM;

  const int mat4   = N_NODES * DIM / 4;        // 3,200,000
  const int gemmG  = N_NODES / 16;             // 6,250 (exact)
  const int scatG  = N_EDGES * 32 / 256;       // 200,000 (exact)
  const int elemG  = mat4 / 256;               // 12,500 (exact)

  // Degree (same for all 3 blocks): zero, count, invert
  zero4_kernel<<<(N_NODES / 4 + 255) / 256, 256, 0, stream>>>((float4*)deg, N_NODES / 4);
  deg_kernel<<<N_EDGES / 256, 256, 0, stream>>>(ei, deg);
  invdeg_kernel<<<(N_NODES + 255) / 256, 256, 0, stream>>>(deg, invd);

  // h0 = relu(x @ in_w + in_b)
  gemm_in_kernel<<<gemmG, 256, 0, stream>>>(x, in_w, in_b, bufA);

  float* bufs[2] = {bufA, bufB};
  for (int blk = 0; blk < 3; ++blk) {
    float* hin  = bufs[blk & 1];
    float* hout = bufs[(blk + 1) & 1];
    zero4_kernel<<<elemG, 256, 0, stream>>>((float4*)agg, mat4);
    zero4_kernel<<<1, 64, 0, stream>>>((float4*)stats, 64);
    scatter_kernel<<<scatG, 256, 0, stream>>>(hin, ei, agg);
    gemm_sage_kernel<<<gemmG, 256, 0, stream>>>(agg, invd, hin,
                                                lw[blk], lb[blk], rw[blk],
                                                hout, stats);
    bn_finalize_kernel<<<1, 128, 0, stream>>>(stats, gg[blk], bt[blk], ss);
    bn_apply_kernel<<<elemG, 256, 0, stream>>>(hout, hin, ss);
  }

  // logits = h @ cls_w + cls_b  (final h lives in bufB after 3 blocks)
  cls_kernel<<<(N_NODES + 255) / 256, 256, 0, stream>>>(bufB, cw, cb, (float*)d_out);
}